// ProjectiveLayer_66675072303463
// MI455X (gfx1250) — compile-verified
//
#include <hip/hip_runtime.h>
#include <hip/hip_bf16.h>
#include <stdint.h>

// Problem constants (match reference: B=64, S=128, N=64, M=1024, W=5)
#define B_    64
#define S_    128
#define N_    64
#define M_    1024
#define W_    5
#define NK    (2 * W_ + 1)          // 11 shifted copies
#define TS    16                    // s-tile width per workgroup (== one 64B line)
#define SPAN  (TS + 2 * W_)         // 26 histogram rows (tile + halo)
#define HSTRIDE (M_ + 1)            // 1025: pad row stride -> conflict-free LDS reads
#define HIST_FLOATS (SPAN * HSTRIDE)            // 26650
#define HIST_PAD    ((HIST_FLOATS + 3) & ~3)    // 26652 -> keeps tile regions 16B aligned
#define NTHREADS 256

// LDS byte offset of a generic pointer that points into shared memory.
__device__ __forceinline__ uint32_t lds_off(const void* p) {
    return (uint32_t)(uintptr_t)(__attribute__((address_space(3))) const void*)p;
}

extern "C" __global__ __launch_bounds__(NTHREADS)
void bloom_window_kernel(const float* __restrict__ wts,   // hash_weights [B,S,N]
                         const int*   __restrict__ mh,    // minhashes    [B,S,N]
                         float*       __restrict__ out)   // [B, 11*M, S]
{
    extern __shared__ float smem[];
    float* hist = smem;                                   // SPAN x HSTRIDE
    int*   mh_t = (int*)  (smem + HIST_PAD);              // SPAN x N  (staged)
    float* w_t  = (float*)(smem + HIST_PAD + SPAN * N_);  // SPAN x N  (staged)

    const int tid  = threadIdx.x;
    const int tile = blockIdx.x;          // 0..7
    const int b    = blockIdx.y;          // 0..63
    const int s0   = tile * TS;

    // ---------------------------------------------------------------
    // 1) Async DMA the input tile (minhashes + weights, 16B chunks)
    //    global -> LDS on the gfx1250 ASYNCcnt path.
    // ---------------------------------------------------------------
    const uint32_t mh_lds0 = lds_off(mh_t);
    const uint32_t w_lds0  = lds_off(w_t);
    // SPAN rows * (16 chunks mh + 16 chunks w) = SPAN*32 b128 transfers
    for (int item = tid; item < SPAN * 32; item += NTHREADS) {
        const int row = item >> 5;
        const int q   = item & 31;
        const int s   = s0 - W_ + row;
        if (s < 0 || s >= S_) continue;   // halo outside sentence: rows stay zero
        const uint32_t rowbytes = (uint32_t)(((b * S_ + s) * N_) * 4);
        if (q < 16) {
            const uint32_t goff = rowbytes + (uint32_t)(q * 16);
            const uint32_t loff = mh_lds0 + (uint32_t)(row * N_ * 4 + q * 16);
            asm volatile("global_load_async_to_lds_b128 %0, %1, %2 offset:0"
                         :: "v"(loff), "v"(goff), "s"(mh) : "memory");
        } else {
            const int qq = q - 16;
            const uint32_t goff = rowbytes + (uint32_t)(qq * 16);
            const uint32_t loff = w_lds0 + (uint32_t)(row * N_ * 4 + qq * 16);
            asm volatile("global_load_async_to_lds_b128 %0, %1, %2 offset:0"
                         :: "v"(loff), "v"(goff), "s"(wts) : "memory");
        }
    }

    // ---------------------------------------------------------------
    // 2) Zero the histograms while the DMA is in flight.
    // ---------------------------------------------------------------
    for (int i = tid; i < HIST_FLOATS; i += NTHREADS) hist[i] = 0.0f;

    asm volatile("s_wait_asynccnt 0" ::: "memory");
    __syncthreads();

    // ---------------------------------------------------------------
    // 3) Weighted scatter histogram: LDS float atomics (ds_add_f32).
    //    M is a power of two -> remainder == AND (inputs non-negative).
    // ---------------------------------------------------------------
    for (int it = tid; it < SPAN * N_; it += NTHREADS) {
        const int row = it >> 6;          // /N_
        const int s   = s0 - W_ + row;
        if (s >= 0 && s < S_) {
            const int bin = mh_t[it] & (M_ - 1);
            atomicAdd(&hist[row * HSTRIDE + bin], w_t[it]);
        }
    }
    __syncthreads();

    // ---------------------------------------------------------------
    // 4) Emit the 11 shifted copies. out[b, k*M+m, s] = counts[b, s-(W-k), m]
    //    which is exactly hist[(s-s0)+k][m]; halo rows are zero already.
    //    Lane mapping: 16 lanes sweep s -> each half-wave writes one full,
    //    aligned 64B cacheline per store. The 369 MB output stream is
    //    write-once with no reuse (~2x L2), so stores are non-temporal
    //    (th:TH_STORE_NT) to keep them from thrashing the 192 MB L2.
    // ---------------------------------------------------------------
    const int ts = tid & (TS - 1);        // 0..15 -> s within tile
    const int mg = tid >> 4;              // 0..15 -> m sub-lane
    float* outb = out + (size_t)b * (NK * M_) * S_ + (size_t)(s0 + ts);
    for (int k = 0; k < NK; ++k) {
        const float* hrow = hist + (ts + k) * HSTRIDE;   // conflict-free (stride 1025)
        float* outk = outb + (size_t)k * M_ * S_;
        #pragma unroll 8
        for (int mb = 0; mb < M_; mb += 16) {
            const int m = mb + mg;
            __builtin_nontemporal_store(hrow[m], &outk[(size_t)m * S_]);
        }
    }
}

extern "C" void kernel_launch(void* const* d_in, const int* in_sizes, int n_in,
                              void* d_out, int out_size, void* d_ws, size_t ws_size,
                              hipStream_t stream) {
    const float* wts = (const float*)d_in[0];   // hash_weights [B,S,N]
    const int*   mh  = (const int*)  d_in[1];   // minhashes    [B,S,N]
    // d_in[2] = M (1024), d_in[3] = W (5): compile-time constants here.
    float* out = (float*)d_out;                 // [B, 11*M, S]

    const dim3 grid(S_ / TS, B_, 1);            // 8 x 64 workgroups
    const dim3 block(NTHREADS, 1, 1);
    const size_t lds_bytes = (size_t)(HIST_PAD + 2 * SPAN * N_) * sizeof(float); // ~120 KB

    bloom_window_kernel<<<grid, block, lds_bytes, stream>>>(wts, mh, out);
}